// GCNModel_31645319036999
// MI455X (gfx1250) — compile-verified
//
#include <hip/hip_runtime.h>
#include <hip/hip_bf16.h>

typedef float v2f __attribute__((ext_vector_type(2)));
typedef float v8f __attribute__((ext_vector_type(8)));

#define NN 50000
#define NE 600000
#define DIM 128
#define NT 5
#define NC 10

// ---------- degree / norm ----------
__global__ void k_init_deg(float* __restrict__ deg) {
    int i = blockIdx.x * blockDim.x + threadIdx.x;
    if (i < NN) deg[i] = 1.0f;          // self-loop contributes 1
}

__global__ void k_accum_deg(const int* __restrict__ dst, float* __restrict__ deg) {
    int i = blockIdx.x * blockDim.x + threadIdx.x;
    if (i < NE) atomicAdd(&deg[dst[i]], 1.0f);
}

__global__ void k_rsqrt_inplace(float* __restrict__ d) {
    int i = blockIdx.x * blockDim.x + threadIdx.x;
    if (i < NN) d[i] = rsqrtf(d[i]);    // deg >= 1 always
}

__global__ void k_zero4(float* __restrict__ p, int n4) {
    int i = blockIdx.x * blockDim.x + threadIdx.x;
    if (i < n4) ((float4*)p)[i] = make_float4(0.f, 0.f, 0.f, 0.f);
}

// ---------- fp32 WMMA GEMM: out[M,128] = A[M,128] @ W[128,128] (+bias)(+relu) ----------
// block = 256 threads (8 waves), computes 16 rows x 128 cols; wave w -> col block 16w.
__launch_bounds__(256)
__global__ void k_gemm128_wmma(const float* __restrict__ A,
                               const float* __restrict__ W,
                               float* __restrict__ out,
                               const float* __restrict__ bias,
                               int relu) {
    __shared__ float aT[16 * 132];     // padded stride 132: conflict-free M-strided reads
    __shared__ float wT[128 * 128];

    const int tid = threadIdx.x;
    const int rowBase = blockIdx.x * 16;

    // load weights 128x128 (4096 float4, 16 per thread)
    const float4* W4 = (const float4*)W;
    float4* wT4 = (float4*)wT;
#pragma unroll
    for (int i = 0; i < 16; ++i) wT4[tid + i * 256] = W4[tid + i * 256];

    // load A tile 16x128 (512 float4, 2 per thread), store with stride 132
#pragma unroll
    for (int i = 0; i < 2; ++i) {
        int j  = tid + i * 256;
        int r  = j >> 5;              // 0..15
        int c4 = j & 31;              // 0..31
        float4 v = ((const float4*)(A + (size_t)(rowBase + r) * DIM))[c4];
        float* d = &aT[r * 132 + c4 * 4];
        d[0] = v.x; d[1] = v.y; d[2] = v.z; d[3] = v.w;
    }
    __syncthreads();

    const int lane = tid & 31;
    const int half = lane >> 4;       // 0: lanes 0-15, 1: lanes 16-31
    const int l15  = lane & 15;
    const int c0   = (tid >> 5) * 16; // wave's column block

    const float* aRow = &aT[l15 * 132];
    v8f acc = {};

#pragma unroll
    for (int k0 = 0; k0 < DIM; k0 += 4) {
        // A 16x4 f32 layout: V0 = K0 | K2 across lane halves, V1 = K1 | K3
        v2f a, b;
        const int ka = k0 + 2 * half;
        a.x = aRow[ka + 0];
        a.y = aRow[ka + 1];
        // B 4x16 f32: rows striped across lanes, analogous half/VGPR split
        b.x = wT[(ka + 0) * DIM + c0 + l15];
        b.y = wT[(ka + 1) * DIM + c0 + l15];
        acc = __builtin_amdgcn_wmma_f32_16x16x4_f32(
            /*neg_a=*/false, a, /*neg_b=*/false, b,
            /*c_mod=*/(short)0, acc, /*reuse_a=*/false, /*reuse_b=*/false);
    }

    const int col = c0 + l15;
    const float bv = bias ? bias[col] : 0.0f;
    // C/D: lanes 0-15 -> rows 0..7 over VGPR 0..7, lanes 16-31 -> rows 8..15
#pragma unroll
    for (int r = 0; r < 8; ++r) {
        int m = r + 8 * half;
        float v = acc[r] + bv;
        if (relu) v = fmaxf(v, 0.0f);
        out[(size_t)(rowBase + m) * DIM + col] = v;
    }
}

// ---------- edge aggregation: one wave32 per edge, float4 per lane ----------
__launch_bounds__(256)
__global__ void k_edge_agg(const float* __restrict__ hin,
                           float* __restrict__ hout,
                           const int* __restrict__ src,
                           const int* __restrict__ dst,
                           const float* __restrict__ dinv) {
    int wave = blockIdx.x * (blockDim.x >> 5) + (threadIdx.x >> 5);
    if (wave >= NE) return;
    int lane = threadIdx.x & 31;
    int s = src[wave];
    int d = dst[wave];
    float nrm = dinv[s] * dinv[d];
    float4 v = ((const float4*)(hin + (size_t)s * DIM))[lane];  // 32*16B = 512B row
    float* q = hout + (size_t)d * DIM + lane * 4;
    atomicAdd(q + 0, v.x * nrm);
    atomicAdd(q + 1, v.y * nrm);
    atomicAdd(q + 2, v.z * nrm);
    atomicAdd(q + 3, v.w * nrm);
}

// ---------- self-loop + bias + relu: agg = relu(agg + hlin*dinv^2 + b) ----------
__global__ void k_self_bias_relu(float* __restrict__ agg,
                                 const float* __restrict__ hlin,
                                 const float* __restrict__ dinv,
                                 const float* __restrict__ bias) {
    int i = blockIdx.x * blockDim.x + threadIdx.x;
    if (i >= NN * DIM) return;
    int n = i >> 7;
    int f = i & 127;
    float di = dinv[n];
    float v = agg[i] + hlin[i] * di * di + bias[f];
    agg[i] = fmaxf(v, 0.0f);
}

// ---------- task heads: y[t,n,c] = h2[n,:] . tw[t,:,c] + tb[t,c] ----------
__launch_bounds__(64)
__global__ void k_heads(const float* __restrict__ h2,
                        const float* __restrict__ tw,
                        const float* __restrict__ tb,
                        float* __restrict__ y) {
    __shared__ float sh[DIM];
    int n = blockIdx.x;
    int tid = threadIdx.x;
    sh[tid]      = h2[(size_t)n * DIM + tid];
    sh[tid + 64] = h2[(size_t)n * DIM + tid + 64];
    __syncthreads();
    if (tid < NT * NC) {
        int t = tid / NC, c = tid % NC;
        const float* wp = tw + (size_t)t * DIM * NC + c;
        float acc = tb[t * NC + c];
#pragma unroll
        for (int f = 0; f < DIM; ++f) acc = fmaf(sh[f], wp[f * NC], acc);
        y[((size_t)t * NN + n) * NC + c] = acc;
    }
}

extern "C" void kernel_launch(void* const* d_in, const int* in_sizes, int n_in,
                              void* d_out, int out_size, void* d_ws, size_t ws_size,
                              hipStream_t stream) {
    const float* x   = (const float*)d_in[0];
    const int*   ei  = (const int*)  d_in[1];
    const float* w1  = (const float*)d_in[2];
    const float* b1  = (const float*)d_in[3];
    const float* w2  = (const float*)d_in[4];
    const float* b2  = (const float*)d_in[5];
    const float* fw1 = (const float*)d_in[6];
    const float* fb1 = (const float*)d_in[7];
    const float* fw2 = (const float*)d_in[8];
    const float* fb2 = (const float*)d_in[9];
    const float* tw  = (const float*)d_in[10];
    const float* tb  = (const float*)d_in[11];

    float* y    = (float*)d_out;                            // [5,50000,10]
    float* feat = y + (size_t)NT * NN * NC;                 // [50000,128]

    float* bufA = (float*)d_ws;                             // N*128
    float* bufB = bufA + (size_t)NN * DIM;                  // N*128
    float* dinv = bufB + (size_t)NN * DIM;                  // N (deg -> dinv in place)

    const int* src = ei;
    const int* dst = ei + NE;

    const int T = 256;
    const int gemmGrid = NN / 16;                           // 3125 (exact)
    const int aggGrid  = (NE + 7) / 8;                      // 8 waves/block
    const int n4       = NN * DIM / 4;

    // degrees -> dinv
    k_init_deg <<<(NN + T - 1) / T, T, 0, stream>>>(dinv);
    k_accum_deg<<<(NE + T - 1) / T, T, 0, stream>>>(dst, dinv);
    k_rsqrt_inplace<<<(NN + T - 1) / T, T, 0, stream>>>(dinv);

    // conv1: bufA = x@w1 ; bufB = relu(scatter(bufA) + self + b1)
    k_gemm128_wmma<<<gemmGrid, T, 0, stream>>>(x, w1, bufA, nullptr, 0);
    k_zero4<<<(n4 + T - 1) / T, T, 0, stream>>>(bufB, n4);
    k_edge_agg<<<aggGrid, T, 0, stream>>>(bufA, bufB, src, dst, dinv);
    k_self_bias_relu<<<(NN * DIM + T - 1) / T, T, 0, stream>>>(bufB, bufA, dinv, b1);

    // conv2: bufA = bufB@w2 ; bufB = relu(scatter(bufA) + self + b2)
    k_gemm128_wmma<<<gemmGrid, T, 0, stream>>>(bufB, w2, bufA, nullptr, 0);
    k_zero4<<<(n4 + T - 1) / T, T, 0, stream>>>(bufB, n4);
    k_edge_agg<<<aggGrid, T, 0, stream>>>(bufA, bufB, src, dst, dinv);
    k_self_bias_relu<<<(NN * DIM + T - 1) / T, T, 0, stream>>>(bufB, bufA, dinv, b2);

    // fc1 -> feature_x (second output), fc2 -> bufA
    k_gemm128_wmma<<<gemmGrid, T, 0, stream>>>(bufB, fw1, feat, fb1, 1);
    k_gemm128_wmma<<<gemmGrid, T, 0, stream>>>(feat, fw2, bufA, fb2, 1);

    // task heads
    k_heads<<<NN, 64, 0, stream>>>(bufA, tw, tb, y);
}